// MeanConv_82008105549850
// MI455X (gfx1250) — compile-verified
//
#include <hip/hip_runtime.h>

typedef float v2f __attribute__((ext_vector_type(2)));
typedef float v8f __attribute__((ext_vector_type(8)));

#define IMG       4096
#define TILE      64
#define HALO      7
#define EXT       (TILE + 2 * HALO)   // 78
#define IN_STRIDE 80
#define CS_STRIDE 81                  // odd stride: conflict-free column reads

__launch_bounds__(256)
__global__ void MeanConv_box7_wmma_kernel(const float* __restrict__ x,
                                          const float* __restrict__ mapf,
                                          float* __restrict__ out)
{
    // 78*80*4 = 24960 B input tile (with replicate halo)
    __shared__ float sIn[EXT * IN_STRIDE];
    // 64*81*4 = 20736 B incremental vertical column sums (64 rows x 78 ext cols)
    __shared__ float sCol[TILE * CS_STRIDE];

    const int tid  = threadIdx.x;
    const int lane = tid & 31;
    const int wave = tid >> 5;
    const int x0   = blockIdx.x * TILE;
    const int y0   = blockIdx.y * TILE;

    // ---- Phase 1: load haloed tile with replicate (edge) clamping ----
    for (int idx = tid; idx < EXT * EXT; idx += 256) {
        const int r  = idx / EXT;
        const int c  = idx - r * EXT;
        int gy = y0 - HALO + r; gy = gy < 0 ? 0 : (gy > IMG - 1 ? IMG - 1 : gy);
        int gx = x0 - HALO + c; gx = gx < 0 ? 0 : (gx > IMG - 1 ? IMG - 1 : gx);
        sIn[r * IN_STRIDE + c] = x[gy * IMG + gx];
    }
    __syncthreads();

    // ---- Phase 2 init: colsum_0 = center row (p = 0 window) ----
    for (int idx = tid; idx < TILE * EXT; idx += 256) {
        const int r = idx / EXT;
        const int c = idx - r * EXT;
        sCol[r * CS_STRIDE + c] = sIn[(r + HALO) * IN_STRIDE + c];
    }
    __syncthreads();

    // Each wave owns two 16x16 subtiles of the 64x64 output tile.
    const int st0 = wave * 2;
    v8f acc[2];
    acc[0] = (v8f){};
    acc[1] = (v8f){};

    const int n_or_m = lane & 15;         // A: row M ; B: col N ; shared formula
    const int khalf  = (lane >> 4) * 2;   // K contribution of lane half (ISA A layout)

#pragma unroll                            // full unroll: wv becomes a literal (no v_div),
    for (int p = 1; p <= HALO; ++p) {     // WMMA chains pipeline across band widths
        // ---- vertical pass: telescoping window update, in place ----
        for (int idx = tid; idx < TILE * EXT; idx += 256) {
            const int r = idx / EXT;
            const int c = idx - r * EXT;
            sCol[r * CS_STRIDE + c] += sIn[(r + HALO - p) * IN_STRIDE + c]
                                     + sIn[(r + HALO + p) * IN_STRIDE + c];
        }
        __syncthreads();

        const float wv = 1.0f / (float)((2 * p + 1) * (2 * p + 1));

        // ---- hoisted band weights: B[k][n] = (|k-7-n| <= p) ? wv : 0 ----
        // identical for both subtiles of this wave (depends on p, t, lane only)
        v2f bw[8];
#pragma unroll
        for (int t = 0; t < 8; ++t) {
            const int k0 = 4 * t + khalf;
            const int d0 = k0 - HALO - n_or_m;
            const int d1 = d0 + 1;
            bw[t].x = (d0 >= -p && d0 <= p) ? wv : 0.0f;
            bw[t].y = (d1 >= -p && d1 <= p) ? wv : 0.0f;
        }

        // ---- horizontal pass as banded matmul on the matrix pipe ----
#pragma unroll
        for (int which = 0; which < 2; ++which) {
            const int st = st0 + which;
            const int sy = (st >> 2) * 16;
            const int sx = (st & 3) * 16;
#pragma unroll
            for (int t = 0; t < 8; ++t) {           // K = 32 (band width 30, zero-padded)
                const int k0 = 4 * t + khalf;       // K for VGPR j=0 in this lane half
                const int k1 = k0 + 1;              // K for VGPR j=1
                int e0 = sx + k0; e0 = e0 < EXT ? e0 : EXT - 1;  // keep reads in-plane
                int e1 = sx + k1; e1 = e1 < EXT ? e1 : EXT - 1;  // (weight is 0 there)
                v2f a;
                a.x = sCol[(sy + n_or_m) * CS_STRIDE + e0];
                a.y = sCol[(sy + n_or_m) * CS_STRIDE + e1];
                acc[which] = __builtin_amdgcn_wmma_f32_16x16x4_f32(
                    /*neg_a=*/false, a, /*neg_b=*/false, bw[t],
                    /*c_mod=*/(short)0, acc[which],
                    /*reuse_a=*/false, /*reuse_b=*/false);
            }
        }
        __syncthreads();  // WAR: WMMA reads of sCol vs next p's update
    }

    // ---- epilogue: /7, mask by map, store ----
    // C/D layout: m = v + 8*(lane>=16), n = lane&15. map is read-once, out is
    // write-once -> non-temporal hints keep L2 for the reused x halos.
    const float inv7 = 1.0f / 7.0f;
    const int   half = lane >> 4;
#pragma unroll
    for (int which = 0; which < 2; ++which) {
        const int st = st0 + which;
        const int sy = (st >> 2) * 16;
        const int sx = (st & 3) * 16;
#pragma unroll
        for (int v = 0; v < 8; ++v) {
            const int mm = v + half * 8;
            const int gy = y0 + sy + mm;
            const int gx = x0 + sx + (lane & 15);
            const long ofs = (long)gy * IMG + gx;
            const float mv = __builtin_nontemporal_load(&mapf[ofs]);
            __builtin_nontemporal_store(acc[which][v] * inv7 * mv, &out[ofs]);
        }
    }
}

extern "C" void kernel_launch(void* const* d_in, const int* in_sizes, int n_in,
                              void* d_out, int out_size, void* d_ws, size_t ws_size,
                              hipStream_t stream) {
    (void)in_sizes; (void)n_in; (void)d_ws; (void)ws_size; (void)out_size;
    const float* x    = (const float*)d_in[0];
    const float* mapf = (const float*)d_in[1];
    float*       out  = (float*)d_out;

    dim3 grid(IMG / TILE, IMG / TILE);   // 64 x 64 tiles
    dim3 block(256);                     // 8 waves per workgroup
    MeanConv_box7_wmma_kernel<<<grid, block, 0, stream>>>(x, mapf, out);
}